// GATv2AnalysisModel_7610682048943
// MI455X (gfx1250) — compile-verified
//
#include <hip/hip_runtime.h>

#define HC 64         // H*C
#define HEADS 4
#define CH 16
#define NEG_SLOPE 0.2f
#define EPSV 1e-16f

typedef float v2f __attribute__((ext_vector_type(2)));
typedef float v8f __attribute__((ext_vector_type(8)));

// ---- monotonic float<->uint encoding for atomic segment-max ------------------
__device__ __forceinline__ unsigned enc_f(float f) {
    unsigned u = __float_as_uint(f);
    return (u & 0x80000000u) ? ~u : (u | 0x80000000u);
}
__device__ __forceinline__ float dec_f(unsigned u) {
    return __uint_as_float((u & 0x80000000u) ? (u & 0x7FFFFFFFu) : ~u);
}

// ---- Y[N,64] = X[N,Fin] @ W[Fin,64] + b  via V_WMMA_F32_16X16X4_F32 ----------
// block = 128 threads (4 waves); wave w -> column tile n0 = 16*w; blockIdx.x -> 16-row tile.
__global__ void __launch_bounds__(128)
gemm16_wmma(const float* __restrict__ X, const float* __restrict__ W,
            const float* __restrict__ bvec, float* __restrict__ Y, int Fin) {
    const int lane = threadIdx.x & 31;
    const int wave = threadIdx.x >> 5;
    const int m0   = blockIdx.x << 4;
    const int n0   = wave << 4;
    const int lm   = lane & 15;
    const int khi  = (lane >> 4) << 1;          // 0 for lanes 0-15, 2 for 16-31

    v8f acc = {};
    const float* xrow = X + (long long)(m0 + lm) * Fin;
    for (int k = 0; k < Fin; k += 4) {
        // A 16x4: lane holds row lm, K = khi, khi+1  (contiguous -> b64 load)
        v2f a = *(const v2f*)(xrow + k + khi);
        // B 4x16: VGPR0 -> K=khi row, VGPR1 -> K=khi+1 row, col = n0+lm
        v2f b;
        b.x = W[(k + khi)     * HC + n0 + lm];
        b.y = W[(k + khi + 1) * HC + n0 + lm];
        acc = __builtin_amdgcn_wmma_f32_16x16x4_f32(false, a, false, b,
                                                    (short)0, acc, false, false);
    }
    const int col = n0 + lm;
    const float bb = bvec[col];
    const int rbase = m0 + ((lane >> 4) << 3);  // C/D: VGPR r -> row r (lo lanes) / r+8 (hi lanes)
    #pragma unroll
    for (int r = 0; r < 8; ++r)
        Y[(long long)(rbase + r) * HC + col] = acc[r] + bb;
}

// ---- pass 1: per-(edge,head) attention score + atomic segment max ------------
__global__ void edge_score(const long long* __restrict__ ei, long long E, long long Etot,
                           const float* __restrict__ xl, const float* __restrict__ xr,
                           const float* __restrict__ att,
                           float* __restrict__ score_out, unsigned* __restrict__ smax_enc) {
    long long t = (long long)blockIdx.x * blockDim.x + threadIdx.x;
    if (t >= Etot * HEADS) return;
    long long e = t >> 2;
    int h = (int)(t & 3);
    long long s, d;
    if (e < E) { s = ei[e]; d = ei[E + e]; } else { s = d = e - E; }
    const float4* pl = (const float4*)(xl + s * HC + h * CH);
    const float4* pr = (const float4*)(xr + d * HC + h * CH);
    const float4* pa = (const float4*)(att + h * CH);
    float sc = 0.f;
    #pragma unroll
    for (int q = 0; q < 4; ++q) {
        float4 a = pl[q], b = pr[q], w = pa[q];
        float m;
        m = a.x + b.x; sc += (m > 0.f ? m : NEG_SLOPE * m) * w.x;
        m = a.y + b.y; sc += (m > 0.f ? m : NEG_SLOPE * m) * w.y;
        m = a.z + b.z; sc += (m > 0.f ? m : NEG_SLOPE * m) * w.z;
        m = a.w + b.w; sc += (m > 0.f ? m : NEG_SLOPE * m) * w.w;
    }
    score_out[t] = sc;                                   // alpha region doubles as scratch
    atomicMax(smax_enc + d * HEADS + h, enc_f(sc));
}

// ---- pass 2: exp(score - max) + atomic segment sum ---------------------------
__global__ void edge_exp(const long long* __restrict__ ei, long long E, long long Etot,
                         const unsigned* __restrict__ smax_enc,
                         float* __restrict__ score_io, float* __restrict__ denom) {
    long long t = (long long)blockIdx.x * blockDim.x + threadIdx.x;
    if (t >= Etot * HEADS) return;
    long long e = t >> 2;
    int h = (int)(t & 3);
    long long d;
    if (e < E) { d = ei[E + e]; } else { d = e - E; }
    float smax = dec_f(smax_enc[d * HEADS + h]);
    float ex = expf(score_io[t] - smax);
    score_io[t] = ex;
    atomicAdd(denom + d * HEADS + h, ex);
}

// ---- pass 3: normalize, emit alpha, scatter-add alpha * xl[src] --------------
__global__ void edge_aggregate(const long long* __restrict__ ei, long long E, long long Etot,
                               const float* __restrict__ xl, const float* __restrict__ denom,
                               float* __restrict__ alpha_io, float* __restrict__ accum) {
    long long t = (long long)blockIdx.x * blockDim.x + threadIdx.x;
    if (t >= Etot * HEADS) return;
    long long e = t >> 2;
    int h = (int)(t & 3);
    long long s, d;
    if (e < E) { s = ei[e]; d = ei[E + e]; } else { s = d = e - E; }
    float a = alpha_io[t] / (denom[d * HEADS + h] + EPSV);
    alpha_io[t] = a;
    const float* pl = xl + s * HC + h * CH;
    float* po = accum + d * HC + h * CH;
    #pragma unroll
    for (int c = 0; c < CH; ++c)
        atomicAdd(po + c, a * pl[c]);
}

// ---- out = relu(accum + bias) ------------------------------------------------
__global__ void finalize_relu(const float* __restrict__ accum, const float* __restrict__ bias,
                              float* __restrict__ out, long long n) {
    long long t = (long long)blockIdx.x * blockDim.x + threadIdx.x;
    if (t >= n) return;
    out[t] = fmaxf(accum[t] + bias[t & (HC - 1)], 0.f);
}

extern "C" void kernel_launch(void* const* d_in, const int* in_sizes, int n_in,
                              void* d_out, int out_size, void* d_ws, size_t ws_size,
                              hipStream_t stream) {
    const float*     x    = (const float*)d_in[0];
    const long long* ei   = (const long long*)d_in[1];
    const float*     Wl1  = (const float*)d_in[2];
    const float*     bl1  = (const float*)d_in[3];
    const float*     Wr1  = (const float*)d_in[4];
    const float*     br1  = (const float*)d_in[5];
    const float*     att1 = (const float*)d_in[6];
    const float*     bias1= (const float*)d_in[7];
    const float*     Wl2  = (const float*)d_in[8];
    const float*     bl2  = (const float*)d_in[9];
    const float*     Wr2  = (const float*)d_in[10];
    const float*     br2  = (const float*)d_in[11];
    const float*     att2 = (const float*)d_in[12];
    const float*     bias2= (const float*)d_in[13];

    const int       Fin1 = in_sizes[2] / HC;                 // 128
    const long long N    = (long long)in_sizes[0] / Fin1;    // 100000
    const long long E    = (long long)in_sizes[1] / 2;       // 1.6M
    const long long Etot = E + N;                            // + self loops

    float* outp   = (float*)d_out;
    float* hfin   = outp;                                    // [N, 64]
    float* alpha1 = outp + N * HC;                           // [Etot, 4]
    float* alpha2 = alpha1 + Etot * HEADS;                   // [Etot, 4]

    // workspace carve-out (smax/denom/accum kept contiguous for one memset)
    char* ws = (char*)d_ws;
    float*    xl    = (float*)ws;    ws += (size_t)N * HC * sizeof(float);
    float*    xr    = (float*)ws;    ws += (size_t)N * HC * sizeof(float);
    float*    h1    = (float*)ws;    ws += (size_t)N * HC * sizeof(float);
    unsigned* smax  = (unsigned*)ws; ws += (size_t)N * HEADS * sizeof(unsigned);
    float*    denom = (float*)ws;    ws += (size_t)N * HEADS * sizeof(float);
    float*    accum = (float*)ws;    ws += (size_t)N * HC * sizeof(float);
    const size_t zero_bytes = (size_t)N * (HEADS + HEADS + HC) * sizeof(float);

    const dim3 gblk(128);
    const dim3 ggrid((unsigned)(N / 16));
    const int  eb = 256;
    const unsigned egrid = (unsigned)((Etot * HEADS + eb - 1) / eb);
    const long long nelem = N * HC;
    const unsigned fgrid = (unsigned)((nelem + eb - 1) / eb);

    // ---------------- layer 1 ----------------
    hipMemsetAsync(smax, 0, zero_bytes, stream);
    gemm16_wmma<<<ggrid, gblk, 0, stream>>>(x, Wl1, bl1, xl, Fin1);
    gemm16_wmma<<<ggrid, gblk, 0, stream>>>(x, Wr1, br1, xr, Fin1);
    edge_score    <<<egrid, eb, 0, stream>>>(ei, E, Etot, xl, xr, att1, alpha1, smax);
    edge_exp      <<<egrid, eb, 0, stream>>>(ei, E, Etot, smax, alpha1, denom);
    edge_aggregate<<<egrid, eb, 0, stream>>>(ei, E, Etot, xl, denom, alpha1, accum);
    finalize_relu <<<fgrid, eb, 0, stream>>>(accum, bias1, h1, nelem);

    // ---------------- layer 2 ----------------
    hipMemsetAsync(smax, 0, zero_bytes, stream);
    gemm16_wmma<<<ggrid, gblk, 0, stream>>>(h1, Wl2, bl2, xl, HC);
    gemm16_wmma<<<ggrid, gblk, 0, stream>>>(h1, Wr2, br2, xr, HC);
    edge_score    <<<egrid, eb, 0, stream>>>(ei, E, Etot, xl, xr, att2, alpha2, smax);
    edge_exp      <<<egrid, eb, 0, stream>>>(ei, E, Etot, smax, alpha2, denom);
    edge_aggregate<<<egrid, eb, 0, stream>>>(ei, E, Etot, xl, denom, alpha2, accum);
    finalize_relu <<<fgrid, eb, 0, stream>>>(accum, bias2, hfin, nelem);
}